// RaceEquivariantBlock_55370718380451
// MI455X (gfx1250) — compile-verified
//
#include <hip/hip_runtime.h>

typedef __attribute__((ext_vector_type(2))) float v2f;
typedef __attribute__((ext_vector_type(4))) float v4f;
typedef __attribute__((ext_vector_type(8))) float v8f;

#define R3  0.57735026918962576451f   /* 1/sqrt(3)  */
#define R2  0.70710678118654752440f   /* 1/sqrt(2)  */
#define R6  0.40824829046386301637f   /* 1/sqrt(6)  */
#define R62 0.81649658092772603273f   /* 2/sqrt(6)  */

// Per-channel term tables: channel c = A(32 x k) * B(k x 32), k <= 3 (padded to 4).
// left sel: 0=s1, 1..3 = v1[0..2]; right sel: 0=s2, 1..3 = v2[0..2]. coef folded into A.
__device__ __constant__ int g_lsel[16][4] = {
  {0,0,0,0},{1,2,3,0},{0,0,0,0},{0,0,0,0},{0,0,0,0},{1,0,0,0},{2,0,0,0},{3,0,0,0},
  {2,3,0,0},{3,1,0,0},{1,2,0,0},{1,3,0,0},{1,2,0,0},{1,2,3,0},{2,3,0,0},{1,3,0,0}};
__device__ __constant__ int g_rsel[16][4] = {
  {0,0,0,0},{1,2,3,0},{1,0,0,0},{2,0,0,0},{3,0,0,0},{0,0,0,0},{0,0,0,0},{0,0,0,0},
  {3,2,0,0},{1,3,0,0},{2,1,0,0},{3,1,0,0},{2,1,0,0},{1,2,3,0},{3,2,0,0},{1,3,0,0}};
__device__ __constant__ float g_coef[16][4] = {
  {1.f,0.f,0.f,0.f},{R3,R3,R3,0.f},
  {1.f,0.f,0.f,0.f},{1.f,0.f,0.f,0.f},{1.f,0.f,0.f,0.f},
  {1.f,0.f,0.f,0.f},{1.f,0.f,0.f,0.f},{1.f,0.f,0.f,0.f},
  {R2,-R2,0.f,0.f},{R2,-R2,0.f,0.f},{R2,-R2,0.f,0.f},
  {R2, R2,0.f,0.f},{R2, R2,0.f,0.f},{-R6,R62,-R6,0.f},{R2,R2,0.f,0.f},{-R2,R2,0.f,0.f}};
// Output placement: idx = base + (u*32+v)*stride + sub
__device__ __constant__ int g_base[16] = {0,1024, 2048,2048,2048, 5120,5120,5120,
                                          8192,8192,8192, 11264,11264,11264,11264,11264};
__device__ __constant__ int g_str[16]  = {1,1, 3,3,3, 3,3,3, 3,3,3, 5,5,5,5,5};
__device__ __constant__ int g_sub[16]  = {0,0, 0,1,2, 0,1,2, 0,1,2, 0,1,2,3,4};

__global__ __launch_bounds__(256, 4)
void race_tp_kernel(const float* __restrict__ x, const float* __restrict__ f,
                    const float* __restrict__ sc, float* __restrict__ out) {
  __shared__ __align__(16) float outbuf[8192];  // half the node output per phase
  __shared__ float xin[128];
  __shared__ float fin[128];

  const int node = blockIdx.x;
  const int tid  = threadIdx.x;
  // Stage this node's features into LDS (coalesced, wave-uniform branch).
  if (tid < 128) xin[tid]       = x[(size_t)node * 128 + tid];
  else           fin[tid - 128] = f[(size_t)node * 128 + (tid - 128)];
  __syncthreads();

  const int lane = tid & 31;
  const int wave = tid >> 5;
  const int half = lane >> 4;       // lane half selects K pair {0,1} vs {2,3}
  const int l15  = lane & 15;       // M (for A) / N (for B/D)
  const int k0   = half * 2, k1 = k0 + 1;
  const size_t nb = (size_t)node * 16384;

  #pragma unroll
  for (int phase = 0; phase < 2; ++phase) {
    const int c   = phase * 8 + wave;           // one channel per wave per phase
    const int ls0 = g_lsel[c][k0], ls1 = g_lsel[c][k1];
    const int rs0 = g_rsel[c][k0], rs1 = g_rsel[c][k1];
    const float q0 = g_coef[c][k0], q1 = g_coef[c][k1];
    const int str  = g_str[c];
    const int bs   = (g_base[c] - phase * 8192) + g_sub[c];
    const int step = 32 * str;                  // dwords between consecutive M rows

    // Build A fragments (16x4 f32: v0 = K0|K2, v1 = K1|K3 by lane half), coef folded in.
    v2f a[2], b[2];
    #pragma unroll
    for (int mt = 0; mt < 2; ++mt) {
      const int u = mt * 16 + l15;
      const float l0 = (ls0 == 0) ? xin[u] : xin[32 + u * 3 + (ls0 - 1)];
      const float l1 = (ls1 == 0) ? xin[u] : xin[32 + u * 3 + (ls1 - 1)];
      a[mt].x = q0 * l0;
      a[mt].y = q1 * l1;
    }
    // B fragments (4x16 f32, mirrored layout: row K0|K2 in v0, K1|K3 in v1).
    #pragma unroll
    for (int nt = 0; nt < 2; ++nt) {
      const int v = nt * 16 + l15;
      b[nt].x = (rs0 == 0) ? fin[v] : fin[32 + v * 3 + (rs0 - 1)];
      b[nt].y = (rs1 == 0) ? fin[v] : fin[32 + v * 3 + (rs1 - 1)];
    }

    // 2x2 tiles of v_wmma_f32_16x16x4_f32 -> 32x32 channel block.
    #pragma unroll
    for (int mt = 0; mt < 2; ++mt) {
      #pragma unroll
      for (int nt = 0; nt < 2; ++nt) {
        v8f d = {0.f, 0.f, 0.f, 0.f, 0.f, 0.f, 0.f, 0.f};
        d = __builtin_amdgcn_wmma_f32_16x16x4_f32(
                false, a[mt], false, b[nt], (short)0, d, false, false);
        // D layout: VGPR r holds M = (mt*16 + half*8 + r), N = nt*16 + l15.
        // Incremental LDS scatter: one mul per tile, one add per store.
        const int uv0 = (mt * 16 + half * 8) * 32 + nt * 16 + l15;
        int idx = bs + uv0 * str;
        #pragma unroll
        for (int r = 0; r < 8; ++r) {
          outbuf[idx] = d[r];
          idx += step;
        }
      }
    }
    __syncthreads();

    // Streaming phase: out = tile + sc, fully coalesced 128-bit traffic (the roofline cost).
    const v4f* sc4 = (const v4f*)(sc + nb + phase * 8192);
    v4f*       o4  = (v4f*)(out + nb + phase * 8192);
    const v4f* t4  = (const v4f*)outbuf;
    #pragma unroll
    for (int it = 0; it < 8; ++it) {
      const int i = it * 256 + tid;
      o4[i] = t4[i] + sc4[i];
    }
    __syncthreads();  // outbuf reused next phase
  }
}

extern "C" void kernel_launch(void* const* d_in, const int* in_sizes, int n_in,
                              void* d_out, int out_size, void* d_ws, size_t ws_size,
                              hipStream_t stream) {
  const float* x  = (const float*)d_in[0];   // x_node_feats (N,128) f32
  const float* f  = (const float*)d_in[1];   // node_feats   (N,128) f32
  const float* sc = (const float*)d_in[2];   // sc           (N,16384) f32
  float* out = (float*)d_out;                // (N,16384) f32
  const int n = in_sizes[0] / 128;           // N = 6000
  race_tp_kernel<<<n, 256, 0, stream>>>(x, f, sc, out);
}